// QPLayer_37108517437990
// MI455X (gfx1250) — compile-verified
//
#include <hip/hip_runtime.h>

typedef float v2f __attribute__((ext_vector_type(2)));
typedef float v8f __attribute__((ext_vector_type(8)));

#define NN 512
#define ITERS 300

__global__ __launch_bounds__(512) void qp_pgd_kernel(const float* __restrict__ mu,
                                                     const float* __restrict__ Sigma,
                                                     float* __restrict__ out)
{
    const int b    = blockIdx.x;
    const int t    = threadIdx.x;      // 0..511
    const int lane = t & 31;
    // Wave index as a *scalar* (SGPR) so tile-loop bounds are wave-uniform:
    // keeps EXEC all-ones around WMMA and enables scalar loop control + unrolling.
    const int waveu = __builtin_amdgcn_readfirstlane(t >> 5);   // 0..15
    const float* __restrict__ S = Sigma + (size_t)b * NN * NN;

    __shared__ float wlds[NN];   // current iterate
    __shared__ float ylds[NN];   // S @ w
    __shared__ float vlds[NN];   // pre-projection point
    __shared__ float slds[NN];   // sort buffer (descending)
    __shared__ float css[NN];    // inclusive cumsum of sorted
    __shared__ float red[NN];    // reduction scratch

    // ---- eta = 1 / (2 * max_row_abs_sum(S) + 1e-6); S symmetric -> column sums (coalesced, one-time)
    float colsum = 0.0f;
    for (int r = 0; r < NN; ++r)
        colsum += fabsf(S[(size_t)r * NN + t]);
    red[t] = colsum;
    __syncthreads();
    for (int off = 256; off > 0; off >>= 1) {
        if (t < off) red[t] = fmaxf(red[t], red[t + off]);
        __syncthreads();
    }
    const float eta   = 1.0f / (2.0f * red[0] + 1e-6f);
    const float mureg = mu[(size_t)b * NN + t];

    wlds[t] = 1.0f / (float)NN;
    __syncthreads();

    // WMMA fragment coordinates (wave32): A 16x4 f32 -> 2 VGPRs
    const int m    = lane & 15;        // row within 16-row block
    const int koff = (lane >> 4) * 2;  // 0 (lanes 0-15) or 2 (lanes 16-31)

    for (int it = 0; it < ITERS; ++it) {
        // ---------- y = S @ w via V_WMMA_F32_16X16X4_F32, block-lower-triangle only ----------
        // Per-sample footprint ~528KB -> 256 samples ~135MB < 192MB L2: iterations 2..300 hit L2.
        // Every wave: (I+1) direct + (31-I) mirrored tile passes = 32 passes per row-block.
        #pragma unroll
        for (int sub = 0; sub < 2; ++sub) {
            const int I  = waveu * 2 + sub;    // owned row-block (scalar)
            const int r0 = I * 16;
            v8f acc = {};

            // direct tiles J = 0..I : A[m][k] = S[r0+m][kg+k], contiguous pair -> b64
            const float* arow = S + (size_t)(r0 + m) * NN + koff;
            const int kend = (I + 1) * 16;     // scalar bound
            #pragma unroll 4
            for (int kg = 0; kg < kend; kg += 4) {
                v2f a = *(const v2f*)(arow + kg);
                __builtin_prefetch(arow + kg + 128, 0, 3);
                v2f bb;
                bb.x = wlds[kg + koff];
                bb.y = wlds[kg + koff + 1];
                acc = __builtin_amdgcn_wmma_f32_16x16x4_f32(false, a, false, bb,
                                                            (short)0, acc, false, false);
            }

            // mirrored below-diagonal tiles I' = I+1..31 (read transposed; still lower triangle):
            // y[r0+m] += sum_r S[rg+r][r0+m] * w[rg+r];  A2[m][k] = S[rg+k][r0+m]
            #pragma unroll 4
            for (int rg = kend; rg < NN; rg += 4) {
                const float* acol = S + (size_t)(rg + koff) * NN + r0 + m;
                v2f a;
                a.x = acol[0];      // lanes 0..15 read 64 contiguous bytes
                a.y = acol[NN];
                v2f bb;
                bb.x = wlds[rg + koff];
                bb.y = wlds[rg + koff + 1];
                acc = __builtin_amdgcn_wmma_f32_16x16x4_f32(false, a, false, bb,
                                                            (short)0, acc, false, false);
            }

            // D: VGPR i <-> row (i + 8*(lane>=16)), col = lane&15. Take col 0.
            if ((lane & 15) == 0) {
                const int base = r0 + (lane >> 4) * 8;
                #pragma unroll
                for (int i2 = 0; i2 < 8; ++i2)
                    ylds[base + i2] = acc[i2];
            }
        }
        __syncthreads();

        // ---------- v = w - eta * (2*y - mu) ----------
        {
            float v = wlds[t] - eta * (2.0f * ylds[t] - mureg);
            vlds[t] = v;
            slds[t] = v;
        }
        __syncthreads();

        // ---------- bitonic sort, descending ----------
        for (int k2 = 2; k2 <= NN; k2 <<= 1) {
            for (int j = k2 >> 1; j > 0; j >>= 1) {
                const int ixj = t ^ j;
                if (ixj > t) {
                    const float a0 = slds[t], a1 = slds[ixj];
                    const bool descBlock = ((t & k2) == 0);
                    if (descBlock ? (a0 < a1) : (a0 > a1)) {
                        slds[t] = a1;
                        slds[ixj] = a0;
                    }
                }
                __syncthreads();
            }
        }

        // ---------- inclusive scan (Hillis-Steele) ----------
        css[t] = slds[t];
        __syncthreads();
        for (int off = 1; off < NN; off <<= 1) {
            const float add = (t >= off) ? css[t - off] : 0.0f;
            __syncthreads();
            css[t] += add;
            __syncthreads();
        }

        // ---------- rho = #{ u_k + (1 - css_k)/k > 0 } ----------
        red[t] = (slds[t] + (1.0f - css[t]) / (float)(t + 1) > 0.0f) ? 1.0f : 0.0f;
        __syncthreads();
        for (int off = 256; off > 0; off >>= 1) {
            if (t < off) red[t] += red[t + off];
            __syncthreads();
        }
        const int   rho   = (int)red[0];
        const float theta = (1.0f - css[rho - 1]) / (float)rho;
        __syncthreads();

        // ---------- w = max(v - theta, 0) ----------
        wlds[t] = fmaxf(vlds[t] - theta, 0.0f);
        __syncthreads();
    }

    out[(size_t)b * NN + t] = wlds[t];
}

extern "C" void kernel_launch(void* const* d_in, const int* in_sizes, int n_in,
                              void* d_out, int out_size, void* d_ws, size_t ws_size,
                              hipStream_t stream) {
    (void)n_in; (void)d_ws; (void)ws_size; (void)out_size;
    const float* mu    = (const float*)d_in[0];
    const float* Sigma = (const float*)d_in[1];
    float*       out   = (float*)d_out;
    const int B = in_sizes[0] / NN;   // 256
    qp_pgd_kernel<<<B, NN, 0, stream>>>(mu, Sigma, out);
}